// SubGraph_15410342658659
// MI455X (gfx1250) — compile-verified
//
#include <hip/hip_runtime.h>
#include <math.h>

typedef float v2f __attribute__((ext_vector_type(2)));
typedef float v8f __attribute__((ext_vector_type(8)));

#define N_NODES 20000
#define N_EDGES 320000
#define HID     64
#define LN_EPS  1e-5f
#define SENT    0x007FFFFFu   /* order-preserving map of -inf */

// ---- order-preserving float <-> uint mapping for atomic max ----
__device__ __forceinline__ unsigned mapf(float f) {
  unsigned u = __float_as_uint(f);
  return (u & 0x80000000u) ? ~u : (u | 0x80000000u);
}
__device__ __forceinline__ float unmapf(unsigned u) {
  unsigned b = (u & 0x80000000u) ? (u & 0x7FFFFFFFu) : ~u;
  return __uint_as_float(b);
}
__device__ __forceinline__ bool finitef(float f) {
  return (__float_as_uint(f) & 0x7F800000u) != 0x7F800000u;
}

// =====================================================================
// Fused per-layer MLP: h = x@W1+b1; LN; ReLU; x2 = h@W2+b2
// One wave32 per 16-node tile. f32 WMMA 16x16x4.
//   A frag (16x4):  lane 0-15 -> M=lane, K = kb+{0,1}; lane 16-31 -> K = kb+{2,3}
//   B frag (4x16):  lane 0-15 -> N=lane, K = kb+{0,1}; lane 16-31 -> K = kb+{2,3}
//   C/D (16x16):    vgpr r, lanes 0-15 -> M=r; lanes 16-31 -> M=r+8; N=lane&15
// Writes x2 into left half of Xn ([row][0..cin)) and seeds right half
// ([row][cin..2cin)) with SENT for the subsequent scatter-max.
// =====================================================================
__global__ __launch_bounds__(32)
void glp_mlp(const float* __restrict__ X, const float* __restrict__ W1,
             const float* __restrict__ b1, const float* __restrict__ gam,
             const float* __restrict__ bet, const float* __restrict__ W2,
             const float* __restrict__ b2, float* __restrict__ Xn,
             int cin, int cshift, int ntiles)
{
  __shared__ float lds[16][HID + 8];
  const int lane = threadIdx.x;
  const int tile = blockIdx.x;
  if (tile >= ntiles) return;                 // wave-uniform; EXEC stays all-1s
  const int m    = lane & 15;
  const int half = lane >> 4;                 // 0 or 1
  const int koff = half << 1;                 // 0 or 2
  const int row0 = tile * 16;
  const size_t cout2 = (size_t)cin * 2;

  // ---------------- GEMM1: h = x @ W1 (M=16, K=cin, N=64) ----------------
  const float* xrow = X + (size_t)(row0 + m) * cin;
  for (int nt = 0; nt < 4; ++nt) {
    const int nb = nt * 16;
    v8f acc = {0.f,0.f,0.f,0.f,0.f,0.f,0.f,0.f};
    for (int kb = 0; kb < cin; kb += 4) {
      v2f a, b;
      a.x = xrow[kb + koff];
      a.y = xrow[kb + koff + 1];
      b.x = W1[(size_t)(kb + koff)     * HID + nb + m];
      b.y = W1[(size_t)(kb + koff + 1) * HID + nb + m];
      acc = __builtin_amdgcn_wmma_f32_16x16x4_f32(
              false, a, false, b, (short)0, acc, false, false);
    }
#pragma unroll
    for (int r = 0; r < 8; ++r) {
      const int mm = r + (half << 3);
      const int nn = nb + m;
      lds[mm][nn] = acc[r] + b1[nn];
    }
  }
  asm volatile("s_wait_dscnt 0" ::: "memory");

  // ---------------- LayerNorm(64) + ReLU, in LDS ----------------
  {
    const int c0 = half * 32;                 // lane pair (m, m+16) splits row m
    float s = 0.f, s2 = 0.f;
#pragma unroll
    for (int c = 0; c < 32; ++c) {
      const float v = lds[m][c0 + c];
      s += v; s2 += v * v;
    }
    s  += __shfl_xor(s, 16, 32);              // wave32 partner reduce
    s2 += __shfl_xor(s2, 16, 32);
    const float mu   = s  * (1.f / HID);
    const float var  = s2 * (1.f / HID) - mu * mu;
    const float rstd = rsqrtf(var + LN_EPS);
#pragma unroll
    for (int c = 0; c < 32; ++c) {
      const int cc = c0 + c;
      float v = (lds[m][cc] - mu) * rstd * gam[cc] + bet[cc];
      lds[m][cc] = fmaxf(v, 0.f);
    }
  }
  asm volatile("s_wait_dscnt 0" ::: "memory");

  // ---------------- GEMM2: x2 = h @ W2 (M=16, K=64, N=cin) ----------------
  for (int nt = 0; nt < (cin >> 4); ++nt) {
    const int nb = nt * 16;
    v8f acc = {0.f,0.f,0.f,0.f,0.f,0.f,0.f,0.f};
    for (int kb = 0; kb < HID; kb += 4) {
      v2f a, b;
      a.x = lds[m][kb + koff];
      a.y = lds[m][kb + koff + 1];
      b.x = W2[(size_t)(kb + koff)     * cin + nb + m];
      b.y = W2[(size_t)(kb + koff + 1) * cin + nb + m];
      acc = __builtin_amdgcn_wmma_f32_16x16x4_f32(
              false, a, false, b, (short)0, acc, false, false);
    }
#pragma unroll
    for (int r = 0; r < 8; ++r) {
      const int mm = r + (half << 3);
      const int nn = nb + m;
      Xn[(size_t)(row0 + mm) * cout2 + nn] = acc[r] + b2[nn];
    }
  }

  // ---------------- seed agg half with mapped(-inf) ----------------
  unsigned* U = (unsigned*)Xn;
  const int per_tile = 16 << cshift;          // 16 * cin
  for (int idx = lane; idx < per_tile; idx += 32) {
    const int r = idx >> cshift;
    const int c = idx & (cin - 1);
    U[(size_t)(row0 + r) * cout2 + cin + c] = SENT;
  }
}

// =====================================================================
// Edge gather + scatter-max: agg[dst] = max(agg[dst], x2[src]) per channel.
// One thread handles one edge x 4 channels (float4 gather, 4 u32 atomics).
// =====================================================================
__global__ __launch_bounds__(256)
void edge_scatter(float* __restrict__ Xn, const long long* __restrict__ src,
                  const long long* __restrict__ dst, int cin, int qshift)
{
  const int gid   = blockIdx.x * blockDim.x + threadIdx.x;
  const int total = N_EDGES << qshift;        // E * cin/4
  if (gid >= total) return;
  const int e  = gid >> qshift;
  const int j4 = gid & ((1 << qshift) - 1);
  const int s  = (int)src[e];
  const int d  = (int)dst[e];
  const size_t cout2 = (size_t)cin * 2;
  const float4 v = *(const float4*)(Xn + (size_t)s * cout2 + j4 * 4);
  unsigned* U = (unsigned*)(Xn + (size_t)d * cout2 + cin + j4 * 4);
  atomicMax(U + 0, mapf(v.x));
  atomicMax(U + 1, mapf(v.y));
  atomicMax(U + 2, mapf(v.z));
  atomicMax(U + 3, mapf(v.w));
}

// Unmap agg half back to float; no-in-edge / non-finite -> 0 (matches ref).
__global__ __launch_bounds__(256)
void agg_finalize(float* __restrict__ Xn, int cin, int cshift)
{
  const int gid   = blockIdx.x * blockDim.x + threadIdx.x;
  const int total = N_NODES << cshift;
  if (gid >= total) return;
  const int r = gid >> cshift;
  const int c = gid & (cin - 1);
  const size_t cout2 = (size_t)cin * 2;
  const unsigned u = ((const unsigned*)Xn)[(size_t)r * cout2 + cin + c];
  float f = unmapf(u);
  if (!finitef(f)) f = 0.f;
  Xn[(size_t)r * cout2 + cin + c] = f;
}

// =====================================================================
// Final column-max over nodes + L2 normalize.
// =====================================================================
__global__ void colmax_init(unsigned* tmp)
{
  tmp[blockIdx.x * blockDim.x + threadIdx.x] = SENT;
}

__global__ __launch_bounds__(512)
void colmax(const float* __restrict__ X, unsigned* __restrict__ tmp, int rpb)
{
  const int col = threadIdx.x;                // 512 columns
  const int r0  = blockIdx.x * rpb;
  int r1 = r0 + rpb; if (r1 > N_NODES) r1 = N_NODES;
  float mx = -INFINITY;
  for (int r = r0; r < r1; ++r)
    mx = fmaxf(mx, X[(size_t)r * 512 + col]);
  atomicMax(&tmp[col], mapf(mx));
}

__global__ __launch_bounds__(512)
void norm_out(const unsigned* __restrict__ tmp, float* __restrict__ out)
{
  __shared__ float red[512];
  const int t = threadIdx.x;
  float f = unmapf(tmp[t]);
  if (!finitef(f)) f = 0.f;
  red[t] = f * f;
  __syncthreads();
  for (int s = 256; s > 0; s >>= 1) {
    if (t < s) red[t] += red[t + s];
    __syncthreads();
  }
  out[t] = f / sqrtf(red[0]);
}

// =====================================================================
extern "C" void kernel_launch(void* const* d_in, const int* in_sizes, int n_in,
                              void* d_out, int out_size, void* d_ws, size_t ws_size,
                              hipStream_t stream)
{
  const float*     x0  = (const float*)d_in[0];
  const long long* ei  = (const long long*)d_in[1];   // [2, E] int64
  const long long* src = ei;
  const long long* dst = ei + N_EDGES;

  float* ws = (float*)d_ws;
  float* X1 = ws;                                   // [N,128]  10.2 MB
  float* X2 = ws + (size_t)N_NODES * 128;           // [N,256]  20.5 MB
  float* X3 = ws + (size_t)N_NODES * 384;           // [N,512]  41.0 MB
  unsigned* tmp = (unsigned*)ws;                    // reuse dead X1 space at end

  const float* Xcur = x0;
  float* Xbufs[3] = {X1, X2, X3};
  const int ntiles = N_NODES / 16;                  // 1250 (exact)

  for (int l = 0; l < 3; ++l) {
    const int cin    = 64 << l;
    const int cshift = 6 + l;
    const float* W1 = (const float*)d_in[2 + 6 * l + 0];
    const float* b1 = (const float*)d_in[2 + 6 * l + 1];
    const float* g  = (const float*)d_in[2 + 6 * l + 2];
    const float* b  = (const float*)d_in[2 + 6 * l + 3];
    const float* W2 = (const float*)d_in[2 + 6 * l + 4];
    const float* b2 = (const float*)d_in[2 + 6 * l + 5];
    float* Xn = Xbufs[l];

    glp_mlp<<<ntiles, 32, 0, stream>>>(Xcur, W1, b1, g, b, W2, b2, Xn,
                                       cin, cshift, ntiles);

    const int qshift = cshift - 2;                  // log2(cin/4)
    const int etotal = N_EDGES << qshift;
    edge_scatter<<<(etotal + 255) / 256, 256, 0, stream>>>(Xn, src, dst, cin, qshift);

    const int ftotal = N_NODES << cshift;
    agg_finalize<<<(ftotal + 255) / 256, 256, 0, stream>>>(Xn, cin, cshift);

    Xcur = Xn;
  }

  colmax_init<<<2, 256, 0, stream>>>(tmp);
  const int rpb = 256;
  colmax<<<(N_NODES + rpb - 1) / rpb, 512, 0, stream>>>(X3, tmp, rpb);
  norm_out<<<1, 512, 0, stream>>>(tmp, (float*)d_out);
}